// GCN_73332271612327
// MI455X (gfx1250) — compile-verified
//
#include <hip/hip_runtime.h>
#include <math.h>

#define NN 100000
#define EE 3200000
#define IN_DIM 512
#define HID 32
#define OUT_DIM 64

typedef __attribute__((ext_vector_type(2))) float v2f;
typedef __attribute__((ext_vector_type(8))) float v8f;

// ---------------------------------------------------------------------------
// init: agg1 = 0, deg = 1 (self-loop contribution)
// ---------------------------------------------------------------------------
__global__ __launch_bounds__(256) void init_kernel(float* __restrict__ agg1,
                                                   float* __restrict__ deg) {
  int gid = blockIdx.x * 256 + threadIdx.x;
  if (gid < NN * HID) agg1[gid] = 0.0f;
  if (gid < NN) deg[gid] = 1.0f;
}

// deg[dst] += 1 over real edges
__global__ __launch_bounds__(256) void deg_kernel(const int* __restrict__ ei,
                                                  float* __restrict__ deg) {
  int gid = blockIdx.x * 256 + threadIdx.x;
  if (gid < EE) atomicAdd(&deg[ei[EE + gid]], 1.0f);
}

// in-place deg -> rsqrt(deg)  (deg >= 1 always, so no zero-guard needed)
__global__ __launch_bounds__(256) void dinv_kernel(float* __restrict__ deg) {
  int gid = blockIdx.x * 256 + threadIdx.x;
  if (gid < NN) deg[gid] = rsqrtf(deg[gid]);
}

// ---------------------------------------------------------------------------
// GEMM1: h0[N,32] = x[N,512] @ W1[512,32]   (exact fp32 WMMA 16x16x4)
// 256 threads = 8 waves; wave w owns rows [base, base+16), both 16-col tiles.
// W1 staged in 64KB LDS, K-pair interleaved:
//   element (k, col) at  w1s[(k>>1)*2*HID + col*2 + (k&1)]
// so each lane's B operand (K, K+1 of one column) is one aligned ds_load_b64.
// ---------------------------------------------------------------------------
__global__ __launch_bounds__(256) void gemm1_kernel(const float* __restrict__ x,
                                                    const float* __restrict__ W1,
                                                    float* __restrict__ h0) {
  __shared__ float w1s[IN_DIM * HID];  // 64 KB
  int tid = threadIdx.x;
  for (int i = tid; i < IN_DIM * HID; i += 256) {
    int k = i >> 5;          // W1 row (K)
    int col = i & 31;        // W1 col (N)
    w1s[(k >> 1) * (2 * HID) + col * 2 + (k & 1)] = W1[i];
  }
  __syncthreads();

  int wave = tid >> 5;
  int lane = tid & 31;
  int m    = lane & 15;   // A row-in-tile / B+C column-in-tile
  int kh   = lane >> 4;   // K-half select (A/B layout) and M-half select (C/D)

  int base = blockIdx.x * 128 + wave * 16;
  int row  = base + m;
  int rowc = row < NN ? row : NN - 1;  // clamp: keep EXEC all-1 for WMMA
  const float* xr = x + (size_t)rowc * IN_DIM;

  v8f c0 = {};
  v8f c1 = {};
  for (int k = 0; k < IN_DIM; k += 4) {
    int kk = k + 2 * kh;                                  // even
    v2f a = *(const v2f*)(xr + kk);                       // A[m][kk], A[m][kk+1]
    const float* bp = w1s + (kk >> 1) * (2 * HID);
    v2f b0  = *(const v2f*)(bp + m * 2);                  // (kk,m),(kk+1,m)
    v2f b1v = *(const v2f*)(bp + (16 + m) * 2);           // (kk,16+m),(kk+1,16+m)
    c0 = __builtin_amdgcn_wmma_f32_16x16x4_f32(false, a, false, b0,  (short)0, c0, false, false);
    c1 = __builtin_amdgcn_wmma_f32_16x16x4_f32(false, a, false, b1v, (short)0, c1, false, false);
  }

  // C/D layout: vgpr r, lanes 0-15 -> M=r, lanes 16-31 -> M=r+8; N = lane%16
#pragma unroll
  for (int r = 0; r < 8; ++r) {
    int rr = base + r + 8 * kh;
    if (rr < NN) {
      h0[(size_t)rr * HID + m]      = c0[r];
      h0[(size_t)rr * HID + 16 + m] = c1[r];
    }
  }
}

// ---------------------------------------------------------------------------
// scatter1: agg1[dst] += h0[src] * dinv[src]*dinv[dst]  (incl. self-loops)
// 8 threads per edge, float4 chunks, f32 global atomics (resolve in L2).
// ---------------------------------------------------------------------------
__global__ __launch_bounds__(256) void scatter1_kernel(const float* __restrict__ h0,
                                                       const int* __restrict__ ei,
                                                       const float* __restrict__ dinv,
                                                       float* __restrict__ agg1) {
  int gid = blockIdx.x * 256 + threadIdx.x;
  if (gid >= (EE + NN) * 8) return;
  int e = gid >> 3;
  int c = gid & 7;
  int src, dst;
  if (e < EE) { src = ei[e]; dst = ei[EE + e]; }
  else        { src = dst = e - EE; }
  float nrm = dinv[src] * dinv[dst];
  float4 v = ((const float4*)(h0 + (size_t)src * HID))[c];
  float* ap = agg1 + (size_t)dst * HID + c * 4;
  atomicAdd(ap + 0, v.x * nrm);
  atomicAdd(ap + 1, v.y * nrm);
  atomicAdd(ap + 2, v.z * nrm);
  atomicAdd(ap + 3, v.w * nrm);
}

// ---------------------------------------------------------------------------
// BN(eval)+bias folded into per-channel affine:  f(u) = u*s + t, then ReLU
//   s = gamma * rsqrt(var+eps);  t = (b1 - mean)*s + beta
// ---------------------------------------------------------------------------
__global__ void prep_kernel(const float* __restrict__ b1,
                            const float* __restrict__ gamma,
                            const float* __restrict__ beta,
                            const float* __restrict__ mean,
                            const float* __restrict__ var,
                            float* __restrict__ st) {
  int i = threadIdx.x;  // exactly 32 threads
  float s = gamma[i] * rsqrtf(var[i] + 1e-5f);
  st[i]       = s;
  st[HID + i] = (b1[i] - mean[i]) * s + beta[i];
}

// ---------------------------------------------------------------------------
// GEMM2: h2[N,64] = relu(agg1*s + t) @ W2[32,64]   (fp32 WMMA 16x16x4)
// 8 waves/block, wave owns 16 rows x 64 cols (4 accumulators), K=32.
// W2 staged K-pair interleaved like W1.
// ---------------------------------------------------------------------------
__global__ __launch_bounds__(256) void gemm2_kernel(const float* __restrict__ agg1,
                                                    const float* __restrict__ W2,
                                                    const float* __restrict__ st,
                                                    float* __restrict__ h2) {
  __shared__ float w2s[HID * OUT_DIM];  // 8 KB
  __shared__ float sst[2 * HID];
  int tid = threadIdx.x;
  for (int i = tid; i < HID * OUT_DIM; i += 256) {
    int k = i >> 6;          // W2 row (K), 0..31
    int col = i & 63;        // W2 col (N), 0..63
    w2s[(k >> 1) * (2 * OUT_DIM) + col * 2 + (k & 1)] = W2[i];
  }
  if (tid < 2 * HID) sst[tid] = st[tid];
  __syncthreads();

  int wave = tid >> 5;
  int lane = tid & 31;
  int m    = lane & 15;
  int kh   = lane >> 4;

  int base = blockIdx.x * 128 + wave * 16;
  int row  = base + m;
  int rowc = row < NN ? row : NN - 1;
  const float* ar = agg1 + (size_t)rowc * HID;

  v8f acc[4] = {{}, {}, {}, {}};
  for (int k = 0; k < HID; k += 4) {
    int kk = k + 2 * kh;                 // even
    v2f raw = *(const v2f*)(ar + kk);
    v2f a;
    a.x = fmaxf(0.0f, raw.x * sst[kk]     + sst[HID + kk]);
    a.y = fmaxf(0.0f, raw.y * sst[kk + 1] + sst[HID + kk + 1]);
    const float* bp = w2s + (kk >> 1) * (2 * OUT_DIM);
#pragma unroll
    for (int j = 0; j < 4; ++j) {
      v2f b = *(const v2f*)(bp + (j * 16 + m) * 2);
      acc[j] = __builtin_amdgcn_wmma_f32_16x16x4_f32(false, a, false, b, (short)0, acc[j], false, false);
    }
  }

#pragma unroll
  for (int r = 0; r < 8; ++r) {
    int rr = base + r + 8 * kh;
    if (rr < NN) {
#pragma unroll
      for (int j = 0; j < 4; ++j)
        h2[(size_t)rr * OUT_DIM + j * 16 + m] = acc[j][r];
    }
  }
}

// out := broadcast b2 (so scatter2 accumulates on top of the bias)
__global__ __launch_bounds__(256) void initout_kernel(const float* __restrict__ b2,
                                                      float* __restrict__ out) {
  int gid = blockIdx.x * 256 + threadIdx.x;
  if (gid < NN * OUT_DIM) out[gid] = b2[gid & (OUT_DIM - 1)];
}

// scatter2: out[dst] += h2[src] * norm   (16 threads/edge, H=64)
__global__ __launch_bounds__(256) void scatter2_kernel(const float* __restrict__ h2,
                                                       const int* __restrict__ ei,
                                                       const float* __restrict__ dinv,
                                                       float* __restrict__ out) {
  long gid = (long)blockIdx.x * 256 + threadIdx.x;
  if (gid >= (long)(EE + NN) * 16) return;
  int e = (int)(gid >> 4);
  int c = (int)(gid & 15);
  int src, dst;
  if (e < EE) { src = ei[e]; dst = ei[EE + e]; }
  else        { src = dst = e - EE; }
  float nrm = dinv[src] * dinv[dst];
  float4 v = ((const float4*)(h2 + (size_t)src * OUT_DIM))[c];
  float* op = out + (size_t)dst * OUT_DIM + c * 4;
  atomicAdd(op + 0, v.x * nrm);
  atomicAdd(op + 1, v.y * nrm);
  atomicAdd(op + 2, v.z * nrm);
  atomicAdd(op + 3, v.w * nrm);
}

// ---------------------------------------------------------------------------
extern "C" void kernel_launch(void* const* d_in, const int* in_sizes, int n_in,
                              void* d_out, int out_size, void* d_ws, size_t ws_size,
                              hipStream_t stream) {
  const float* x     = (const float*)d_in[0];
  const int*   ei    = (const int*)d_in[1];
  const float* W1    = (const float*)d_in[2];
  const float* b1    = (const float*)d_in[3];
  const float* W2    = (const float*)d_in[4];
  const float* b2    = (const float*)d_in[5];
  const float* gamma = (const float*)d_in[6];
  const float* beta  = (const float*)d_in[7];
  const float* rmean = (const float*)d_in[8];
  const float* rvar  = (const float*)d_in[9];
  float* out = (float*)d_out;

  // workspace layout (floats)
  float* wsf  = (float*)d_ws;
  size_t o    = 0;
  float* deg  = wsf + o; o += 102400;               // N, rounded for alignment
  float* h0   = wsf + o; o += (size_t)NN * HID;     // 3.2M
  float* agg1 = wsf + o; o += (size_t)NN * HID;     // 3.2M
  float* h2   = wsf + o; o += (size_t)NN * OUT_DIM; // 6.4M
  float* st   = wsf + o; o += 64;

  int blkInit  = (NN * HID + 255) / 256;
  int blkE     = (EE + 255) / 256;
  int blkN     = (NN + 255) / 256;
  int blkGemm  = (NN + 127) / 128;
  int blkSc1   = (int)(((long)(EE + NN) * 8 + 255) / 256);
  int blkOut   = (NN * OUT_DIM + 255) / 256;
  int blkSc2   = (int)(((long)(EE + NN) * 16 + 255) / 256);

  init_kernel    <<<blkInit, 256, 0, stream>>>(agg1, deg);
  deg_kernel     <<<blkE,    256, 0, stream>>>(ei, deg);
  dinv_kernel    <<<blkN,    256, 0, stream>>>(deg);           // deg -> dinv in place
  gemm1_kernel   <<<blkGemm, 256, 0, stream>>>(x, W1, h0);
  scatter1_kernel<<<blkSc1,  256, 0, stream>>>(h0, ei, deg, agg1);
  prep_kernel    <<<1,        32, 0, stream>>>(b1, gamma, beta, rmean, rvar, st);
  gemm2_kernel   <<<blkGemm, 256, 0, stream>>>(agg1, W2, st, h2);
  initout_kernel <<<blkOut,  256, 0, stream>>>(b2, out);
  scatter2_kernel<<<blkSc2,  256, 0, stream>>>(h2, ei, deg, out);
}